// MyLayer_40364102648123
// MI455X (gfx1250) — compile-verified
//
#include <hip/hip_runtime.h>

// Problem dims (match reference)
#define B_ 4
#define T_ 1024
#define D_ 256
#define O_ 256

// Tiling
#define OB   128            // o-columns per block (8 waves x 16)
#define CH   16             // k-rows per TDM chunk (8 KB tile)
#define NCH  (D_ / CH)      // 16 chunks
#define WP   136            // LDS row stride for W tile (128 + 8 pad -> bank-clean)
#define AP   260            // LDS row stride for A (256 + 4 pad -> bank-clean)
#define NBUF 3              // triple buffer

typedef __attribute__((ext_vector_type(2))) float        v2f;
typedef __attribute__((ext_vector_type(8))) float        v8f;
typedef __attribute__((ext_vector_type(4))) unsigned int v4u;
typedef __attribute__((ext_vector_type(4))) int          v4i;
typedef __attribute__((ext_vector_type(8))) int          v8i;

#if __has_builtin(__builtin_amdgcn_tensor_load_to_lds)
#  define HAVE_TDM 1
#else
#  define HAVE_TDM 0
#endif

// ---------------------------------------------------------------------------
// Kernel 1: cumulative mean along T (8 MB traffic; coalesced across d).
// ---------------------------------------------------------------------------
__global__ void cummean_k(const float* __restrict__ x, float* __restrict__ cm) {
    const int idx = blockIdx.x * blockDim.x + threadIdx.x;   // 0 .. B*D-1
    const int b = idx / D_;
    const int d = idx % D_;
    const float* xp = x  + (size_t)b * T_ * D_ + d;
    float*       cp = cm + (size_t)b * T_ * D_ + d;
    float acc = 0.0f;
    for (int t = 0; t < T_; ++t) {
        acc += xp[(size_t)t * D_];
        cp[(size_t)t * D_] = acc / (float)(t + 1);
    }
}

// ---------------------------------------------------------------------------
// TDM issue: load a 16x128 f32 tile of W (row stride O_=256 floats) into LDS
// at byte offset lds_off, with LDS padding of 8 DWORDs after every 128 DWORDs
// (-> LDS row stride 136 floats).  D# layout per CDNA5 ISA 8.3/8.4.
// ---------------------------------------------------------------------------
#if HAVE_TDM
__device__ __forceinline__ unsigned lds_byte_off(const void* p) {
    // Generic pointers to LDS carry the LDS byte offset in their low 32 bits.
    return (unsigned)(unsigned long long)(uintptr_t)p;
}

__device__ __forceinline__ void tdm_issue_w_tile(const float* gsrc, unsigned lds_off) {
    const unsigned long long ga = (unsigned long long)(uintptr_t)gsrc;

    v4u g0;
    g0.x = 1u;                                          // count=1 (valid), user mode
    g0.y = lds_off;                                     // lds_addr (bytes)
    g0.z = (unsigned)ga;                                // global_addr[31:0]
    g0.w = (unsigned)((ga >> 32) & 0x1FFFFFFull)        // global_addr[56:32]
         | (2u << 30);                                  // type = 2 ("image")

    v8i g1;
    g1[0] = (int)((2u << 16)                            // data_size = 4 bytes
                | (1u << 20)                            // pad_enable
                | (6u << 22)                            // pad_interval: 128 DWORDs
                | (7u << 25));                          // pad_amount:   8 DWORDs
    g1[1] = (int)(((unsigned)OB) << 16);                // tensor_dim0 = 128
    g1[2] = (int)(((unsigned)CH) << 16);                // tensor_dim1 = 16
    g1[3] = (int)(((unsigned)OB) << 16);                // tile_dim0   = 128
    g1[4] = (int)(unsigned)CH;                          // tile_dim1 = 16, tile_dim2 = 0
    g1[5] = (int)(unsigned)O_;                          // tensor_dim0_stride = 256
    g1[6] = 0;
    g1[7] = 0;

    v4i gz; gz.x = 0; gz.y = 0; gz.z = 0; gz.w = 0;     // groups 2/3 unused (2D)

#if __has_include(<hip/amd_detail/amd_gfx1250_TDM.h>)
    // amdgpu-toolchain (clang-23): 6-arg form with extra group.
    v8i gz8; gz8[0]=0; gz8[1]=0; gz8[2]=0; gz8[3]=0; gz8[4]=0; gz8[5]=0; gz8[6]=0; gz8[7]=0;
    __builtin_amdgcn_tensor_load_to_lds(g0, g1, gz, gz, gz8, 0);
#else
    // ROCm 7.2 (clang-22): 5-arg form.
    __builtin_amdgcn_tensor_load_to_lds(g0, g1, gz, gz, 0);
#endif
}
#endif  // HAVE_TDM

// ---------------------------------------------------------------------------
// Kernel 2: per-t GEMM  out[:, t, :] = cm[:, t, :] @ W[t]   (M=4,K=256,N=256)
// grid = (2, T), block = 256 = 8 wave32 waves; wave w owns o-tile w within the
// block's 128-wide O slab.  W is streamed HBM -> LDS by the Tensor Data Mover
// (triple-buffered, TENSORcnt-tracked, one barrier per 16-row chunk), and
// consumed by v_wmma_f32_16x16x4_f32 from LDS.
// ---------------------------------------------------------------------------
__global__ __launch_bounds__(256)
void cmgemm_wmma_k(const float* __restrict__ cm,
                   const float* __restrict__ W,
                   float* __restrict__ out) {
    const int t     = blockIdx.y;
    const int lane  = threadIdx.x & 31;
    const int wave  = threadIdx.x >> 5;          // 0..7
    const int o_blk = blockIdx.x * OB;

    __shared__ float As[16 * AP];                // A: 16 rows (4 real + 12 zero)
    __shared__ float Wl[NBUF * CH * WP];         // W tiles, triple buffered

    const float* Wt = W + (size_t)t * D_ * O_ + o_blk;

#if HAVE_TDM
    // Prologue: wave 0 launches the first two chunk DMAs (uniform branch).
    if (wave == 0) {
        tdm_issue_w_tile(Wt,                lds_byte_off(&Wl[0]));
        tdm_issue_w_tile(Wt + CH * O_,      lds_byte_off(&Wl[CH * WP]));
    }
#endif

    // Stage A = cm[:, t, :]: zero-pad to 16 rows, then fill rows 0..3.
    for (int i = threadIdx.x; i < 16 * AP; i += 256) As[i] = 0.0f;
    __syncthreads();
    #pragma unroll
    for (int j = 0; j < B_; ++j) {
        As[j * AP + threadIdx.x] =
            cm[(size_t)j * T_ * D_ + (size_t)t * D_ + threadIdx.x];
    }
    // (published by the first in-loop barrier)

    const int m  = lane & 15;                    // A row
    const int hi = lane >> 4;                    // K half-select
    const int n  = lane & 15;                    // B/C column
    const int wc = wave << 4;                    // wave's column base in slab

    v8f acc = {};

    for (int c = 0; c < NCH; ++c) {
#if HAVE_TDM
        if (wave == 0) {
            // chunks issued so far: 0..min(c+1, NCH-1); require chunk c done.
            if (c + 1 < NCH) __builtin_amdgcn_s_wait_tensorcnt(1);
            else             __builtin_amdgcn_s_wait_tensorcnt(0);
        }
        __syncthreads();                         // chunk c visible; buf (c-1)%3 free
        const float* Wb = &Wl[(c % NBUF) * CH * WP];
#else
        __syncthreads();
        {   // Fallback: cooperative synchronous copy, single buffer.
            const float* src = Wt + (size_t)c * CH * O_;
            for (int i = threadIdx.x; i < CH * OB; i += 256) {
                const int r = i >> 7, col = i & (OB - 1);
                Wl[r * WP + col] =
                    __builtin_nontemporal_load(src + (size_t)r * O_ + col);
            }
        }
        __syncthreads();
        const float* Wb = &Wl[0];
#endif

        #pragma unroll
        for (int s = 0; s < CH / 4; ++s) {
            const int kr = 4 * s + 2 * hi;       // row within chunk
            v2f a;                               // unconditional ds_load_b64
            a.x = As[m * AP + c * CH + kr];
            a.y = As[m * AP + c * CH + kr + 1];
            v2f bm;                              // bank-clean (stride 136)
            bm.x = Wb[(kr)     * WP + wc + n];
            bm.y = Wb[(kr + 1) * WP + wc + n];
            acc = __builtin_amdgcn_wmma_f32_16x16x4_f32(
                false, a, false, bm, (short)0, acc, false, false);
        }

#if HAVE_TDM
        // Issue chunk c+2 into buf (c+2)%3 == (c-1)%3; the barrier above proved
        // every wave finished computing chunk c-1 from that buffer.
        if (wave == 0 && c + 2 < NCH) {
            tdm_issue_w_tile(Wt + (size_t)(c + 2) * CH * O_,
                             lds_byte_off(&Wl[((c + 2) % NBUF) * CH * WP]));
        }
#endif
    }

    // Store rows M=0..3 (batch): VGPRs 0..3, lanes 0..15.
    if (hi == 0) {
        #pragma unroll
        for (int vr = 0; vr < B_; ++vr) {
            __builtin_nontemporal_store(
                acc[vr],
                out + (size_t)vr * T_ * O_ + (size_t)t * O_ + o_blk + wc + n);
        }
    }
}

// ---------------------------------------------------------------------------
// Launch: cm (B*T*D floats = 4 MB) lives in d_ws scratch.
// ---------------------------------------------------------------------------
extern "C" void kernel_launch(void* const* d_in, const int* in_sizes, int n_in,
                              void* d_out, int out_size, void* d_ws, size_t ws_size,
                              hipStream_t stream) {
    const float* x = (const float*)d_in[0];   // (B, T, D) f32
    const float* W = (const float*)d_in[1];   // (T, D, O) f32
    float* out = (float*)d_out;               // (B, T, O) f32
    float* cm  = (float*)d_ws;                // needs B*T*D*4 = 4 MB scratch

    cummean_k<<<dim3((B_ * D_) / 256), dim3(256), 0, stream>>>(x, cm);
    cmgemm_wmma_k<<<dim3(2, T_), dim3(256), 0, stream>>>(cm, W, out);
}